// SS_GDE_28252294873160
// MI455X (gfx1250) — compile-verified
//
#include <hip/hip_runtime.h>
#include <hip/hip_bf16.h>

typedef __attribute__((ext_vector_type(2))) float v2f;
typedef __attribute__((ext_vector_type(8))) float v8f;
typedef __attribute__((ext_vector_type(4))) int   v4i_t;

#define EPSF 1e-6f

// CDNA5 async global->LDS copy support (guarded so absence cannot break compile)
#if __has_builtin(__builtin_amdgcn_global_load_async_to_lds_b32) && \
    __has_builtin(__builtin_amdgcn_global_load_async_to_lds_b128) && \
    __has_builtin(__builtin_amdgcn_s_wait_asynccnt)
#define HAVE_ASYNC_LDS 1
// clang prototypes (from diagnostics): arg0 = typed global (addrspace(1)) ptr,
// arg1 = typed LDS (addrspace(3)) ptr, then imm offset, imm cpol.
#define GASI(p)  ((__attribute__((address_space(1))) int*)(p))
#define LASI(p)  ((__attribute__((address_space(3))) int*)(p))
#define GASI4(p) ((__attribute__((address_space(1))) v4i_t*)(p))
#define LASI4(p) ((__attribute__((address_space(3))) v4i_t*)(p))
#else
#define HAVE_ASYNC_LDS 0
#endif

// ---------------------------------------------------------------------------
// WMMA helpers: fp32 16x16x4 (D = A*B + C), per CDNA5 VGPR layouts.
// A frag (16x4): lane 0-15 -> M=lane, K={k0,k0+1}; lane 16-31 -> K={k0+2,k0+3}
// B frag (4x16): lane 0-15 -> N=lane, K={k0,k0+1}; lane 16-31 -> K={k0+2,k0+3}
// C/D frag: vgpr v, lanes 0-15 -> row m0+v, lanes 16-31 -> row m0+8+v, col=lane&15
// ---------------------------------------------------------------------------
__device__ __forceinline__ v8f wmma_tile_f32(const float* A, int lda,
                                             const float* B, int ldb,
                                             int m0, int n0, int K,
                                             v8f acc, bool b_trans)
{
    const int lane = threadIdx.x & 31;
    const int half = lane >> 4;
    const int rc   = lane & 15;
    for (int k0 = 0; k0 < K; k0 += 4) {
        const int ka = k0 + 2 * half;
        v2f a, b;
        a.x = A[(m0 + rc) * lda + ka];
        a.y = A[(m0 + rc) * lda + ka + 1];
        if (b_trans) {
            b.x = B[(n0 + rc) * ldb + ka];
            b.y = B[(n0 + rc) * ldb + ka + 1];
        } else {
            b.x = B[(ka) * ldb + n0 + rc];
            b.y = B[(ka + 1) * ldb + n0 + rc];
        }
        acc = __builtin_amdgcn_wmma_f32_16x16x4_f32(false, a, false, b,
                                                    (short)0, acc, false, false);
    }
    return acc;
}

__device__ __forceinline__ void store_tile_f32(float* C, int ldc, int m0, int n0,
                                               v8f acc, const float* bias, int relu)
{
    const int lane = threadIdx.x & 31;
    const int half = lane >> 4;
    const int col  = lane & 15;
#pragma unroll
    for (int v = 0; v < 8; ++v) {
        float x = acc[v];
        if (bias) x += bias[n0 + col];
        if (relu) x = fmaxf(x, 0.0f);
        C[(m0 + v + 8 * half) * ldc + n0 + col] = x;
    }
}

__device__ __forceinline__ v8f zero8() {
    v8f z = {0.f, 0.f, 0.f, 0.f, 0.f, 0.f, 0.f, 0.f};
    return z;
}

// ---------------------------------------------------------------------------
// threefry2x32 with key = [0, 42]  (jax.random.key(42)), matches JAX exactly
// ---------------------------------------------------------------------------
__device__ __forceinline__ unsigned rotl32(unsigned x, unsigned r) {
    return (x << r) | (x >> (32u - r));
}
__device__ __forceinline__ void threefry_0_42(unsigned c0, unsigned c1,
                                              unsigned& o0, unsigned& o1)
{
    const unsigned k0 = 0u, k1 = 42u;
    const unsigned ks[3] = {k0, k1, 0x1BD11BDAu ^ k0 ^ k1};
    const unsigned rotA[4] = {13u, 15u, 26u, 6u};
    const unsigned rotB[4] = {17u, 29u, 16u, 24u};
    unsigned x0 = c0 + ks[0];
    unsigned x1 = c1 + ks[1];
#pragma unroll
    for (int i = 0; i < 5; ++i) {
        const unsigned* rr = (i & 1) ? rotB : rotA;
#pragma unroll
        for (int j = 0; j < 4; ++j) {
            x0 += x1;
            x1 = rotl32(x1, rr[j]);
            x1 ^= x0;
        }
        x0 += ks[(i + 1) % 3];
        x1 += ks[(i + 2) % 3] + (unsigned)(i + 1);
    }
    o0 = x0; o1 = x1;
}

// ---------------------------------------------------------------------------
// K1/K2: 3-layer GCN, one workgroup per graph (64 nodes).
//   h1 = relu(adj @ (X @ W1) + b1)   (64x128)  -- computed in 2 column halves
//   h2 = relu(adj @ (h1 @ W2) + b2)  (64x64)
//   out = adj @ (h2 @ W3) + b3       (64x32)
// ---------------------------------------------------------------------------
__global__ __launch_bounds__(256) void gcn_kernel(
    const float* __restrict__ feat, const float* __restrict__ adj,
    const float* __restrict__ W1, const float* __restrict__ b1,
    const float* __restrict__ W2, const float* __restrict__ b2,
    const float* __restrict__ W3, const float* __restrict__ b3,
    float* __restrict__ out)
{
    __shared__ __attribute__((aligned(16))) float adjS[64 * 64];
    __shared__ __attribute__((aligned(16))) float bufX[64 * 64];   // X, then Z/H, then H2
    __shared__ __attribute__((aligned(16))) float bufT[64 * 128];  // T; later T2, T3

    const int g    = blockIdx.x;
    const int tid  = threadIdx.x;
    const int wave = __builtin_amdgcn_readfirstlane(tid >> 5);

#if HAVE_ASYNC_LDS
    // VGPR-bypassing async copy, tracked by ASYNCcnt (16B aligned: stride 64)
    for (int i4 = tid * 4; i4 < 4096; i4 += 1024) {
        __builtin_amdgcn_global_load_async_to_lds_b128(GASI4(adj + g * 4096 + i4),
                                                       LASI4(&adjS[i4]), 0, 0);
        __builtin_amdgcn_global_load_async_to_lds_b128(GASI4(feat + g * 4096 + i4),
                                                       LASI4(&bufX[i4]), 0, 0);
    }
    __builtin_amdgcn_s_wait_asynccnt(0);
#else
    for (int i = tid; i < 4096; i += 256) {
        adjS[i] = adj[g * 4096 + i];
        bufX[i] = feat[g * 4096 + i];
    }
#endif
    __syncthreads();

    // T = X @ W1 : (64x128), K=64 -> 32 tiles, 4 per wave
    for (int t = wave; t < 32; t += 8) {
        int m0 = (t >> 3) << 4, n0 = (t & 7) << 4;
        v8f acc = zero8();
        acc = wmma_tile_f32(bufX, 64, W1, 128, m0, n0, 64, acc, false);
        store_tile_f32(bufT, 128, m0, n0, acc, nullptr, 0);
    }
    __syncthreads();

    // T2 = sum_h  relu(adj @ T[:,h] + b1[h]) @ W2[h,:]   accumulated in registers
    v8f t2acc[2] = {zero8(), zero8()};
    for (int h = 0; h < 2; ++h) {
        __syncthreads();
        for (int t = wave; t < 16; t += 8) {
            int m0 = (t >> 2) << 4, n0 = (t & 3) << 4;
            v8f acc = zero8();
            acc = wmma_tile_f32(adjS, 64, bufT + h * 64, 128, m0, n0, 64, acc, false);
            store_tile_f32(bufX, 64, m0, n0, acc, b1 + h * 64, 1);
        }
        __syncthreads();
        for (int ti = 0; ti < 2; ++ti) {
            int t = wave + ti * 8;
            int m0 = (t >> 2) << 4, n0 = (t & 3) << 4;
            t2acc[ti] = wmma_tile_f32(bufX, 64, W2 + (h * 64) * 64, 64,
                                      m0, n0, 64, t2acc[ti], false);
        }
    }
    __syncthreads();
    for (int ti = 0; ti < 2; ++ti) {
        int t = wave + ti * 8;
        int m0 = (t >> 2) << 4, n0 = (t & 3) << 4;
        store_tile_f32(bufT, 64, m0, n0, t2acc[ti], nullptr, 0);
    }
    __syncthreads();

    // H2 = relu(adj @ T2 + b2)  (64x64)
    for (int t = wave; t < 16; t += 8) {
        int m0 = (t >> 2) << 4, n0 = (t & 3) << 4;
        v8f acc = zero8();
        acc = wmma_tile_f32(adjS, 64, bufT, 64, m0, n0, 64, acc, false);
        store_tile_f32(bufX, 64, m0, n0, acc, b2, 1);
    }
    __syncthreads();

    // T3 = H2 @ W3  (64x32)
    float* T3 = bufT + 4096;
    for (int t = wave; t < 8; t += 8) {
        int m0 = (t >> 1) << 4, n0 = (t & 1) << 4;
        v8f acc = zero8();
        acc = wmma_tile_f32(bufX, 64, W3, 32, m0, n0, 64, acc, false);
        store_tile_f32(T3, 32, m0, n0, acc, nullptr, 0);
    }
    __syncthreads();

    // out = adj @ T3 + b3  (64x32) -> global
    for (int t = wave; t < 8; t += 8) {
        int m0 = (t >> 1) << 4, n0 = (t & 1) << 4;
        v8f acc = zero8();
        acc = wmma_tile_f32(adjS, 64, T3, 32, m0, n0, 64, acc, false);
        store_tile_f32(out + g * 2048, 32, m0, n0, acc, b3, 0);
    }
}

// ---------------------------------------------------------------------------
// K3: attention (folded cos-einsum), Bernoulli sampling (threefry), d_samp, kl
//   T[f] = sum_b mlp2_w[b] * (sum_n yn[b,n,f]*mlp_w[n])
//   attn[d,m] = sigmoid(sum_f dn[d,m,f]*T[f] + mlp_b*sum(mlp2_w) + mlp2_b)
// ---------------------------------------------------------------------------
__global__ __launch_bounds__(256) void attn_kernel(
    const float* __restrict__ xenc, const float* __restrict__ denc,
    const float* __restrict__ mlp_w, const float* __restrict__ mlp_b,
    const float* __restrict__ mlp2_w, const float* __restrict__ mlp2_b,
    float* __restrict__ dsamp, float* __restrict__ klout)
{
    __shared__ float S[1024];    // [b][f]
    __shared__ float Tf[32];
    __shared__ float ndn[960];   // normd [d][f]
    __shared__ float c0s;
    __shared__ float red[256];
    const int tid = threadIdx.x;

    for (int p = tid; p < 1024; p += 256) {
        int b = p >> 5, f = p & 31;
        float ss = 0.f, dot = 0.f;
        for (int n = 0; n < 64; ++n) {
            float v = xenc[(b * 64 + n) * 32 + f];
            ss  += v * v;
            dot += v * mlp_w[n];
        }
        S[p] = dot / (sqrtf(ss) + 1e-12f);
    }
    for (int p = tid; p < 960; p += 256) {
        int d = p >> 5, f = p & 31;
        float ss = 0.f;
        for (int m = 0; m < 64; ++m) {
            float v = denc[(d * 64 + m) * 32 + f];
            ss += v * v;
        }
        ndn[p] = sqrtf(ss) + 1e-12f;
    }
    __syncthreads();
    if (tid < 32) {
        float s = 0.f;
        for (int b = 0; b < 32; ++b) s += S[b * 32 + tid] * mlp2_w[b];
        Tf[tid] = s;
    }
    if (tid == 0) {
        float s = 0.f;
        for (int b = 0; b < 32; ++b) s += mlp2_w[b];
        c0s = mlp_b[0] * s + mlp2_b[0];
    }
    __syncthreads();

    float klacc = 0.f;
    for (int p = tid; p < 1920; p += 256) {
        int d = p >> 6, m = p & 63;
        float pr = 0.f;
        for (int f = 0; f < 32; ++f)
            pr += (denc[(d * 64 + m) * 32 + f] / ndn[d * 32 + f]) * Tf[f];
        pr += c0s;
        float a = 1.f / (1.f + expf(-pr));
        // exact JAX uniform(key(42), (30,64)) bits
        unsigned i0 = (p < 960) ? (unsigned)p : (unsigned)(p - 960);
        unsigned i1 = i0 + 960u;
        unsigned o0, o1;
        threefry_0_42(i0, i1, o0, o1);
        unsigned bits = (p < 960) ? o0 : o1;
        float u = __uint_as_float((bits >> 9) | 0x3f800000u) - 1.0f;
        float bern = (u < a) ? 1.f : 0.f;
        for (int f = 0; f < 32; ++f)
            dsamp[(d * 64 + m) * 32 + f] = bern * denc[(d * 64 + m) * 32 + f];
        klacc += a * logf(a * 2.f) + (1.f - a) * logf((1.f - a) * 2.f);
    }
    red[tid] = klacc;
    __syncthreads();
    for (int s = 128; s > 0; s >>= 1) {
        if (tid < s) red[tid] += red[tid + s];
        __syncthreads();
    }
    if (tid == 0) *klout = red[0];
}

// ---------------------------------------------------------------------------
// K4: fused per-(b,d) tile:  M (WMMA) -> [per lambda] exp/mask -> Sinkhorn(10)
//     -> G minmax-normalize -> (G*M) @ dist_para  -> gd[l*B+b, d, 0:32]
// ---------------------------------------------------------------------------
__global__ __launch_bounds__(256) void fuse_kernel(
    const float* __restrict__ xenc, const float* __restrict__ dsamp,
    const int* __restrict__ num_node, const int* __restrict__ dict_nnode,
    const float* __restrict__ dist_para, float* __restrict__ gd)
{
    __shared__ float xe[64 * 33];
    __shared__ float ds[64 * 33];
    __shared__ float Mld[64 * 65];
    __shared__ float Kld[64 * 65];
    __shared__ float x2[64], y2[64], wiL[64], wjL[64], rmk[64], cmk[64];
    __shared__ float uiL[64], viL[64];
    __shared__ float part[4 * 64];
    __shared__ float pmin[4 * 64], pmax[4 * 64];
    __shared__ float rmin[64], rmax[64];
    __shared__ float gpart[256];

    const float LAM[8] = {0.01f, 0.1f, 0.5f, 1.0f, 3.0f, 5.0f, 10.0f, 20.0f};
    const int b = blockIdx.x & 31;
    const int d = blockIdx.x >> 5;
    const int tid = threadIdx.x;
    const int wave = __builtin_amdgcn_readfirstlane(tid >> 5);

#if HAVE_ASYNC_LDS
    for (int i = tid; i < 2048; i += 256) {
        int n = i >> 5, f = i & 31;
        __builtin_amdgcn_global_load_async_to_lds_b32(GASI(xenc + b * 2048 + i),
                                                      LASI(&xe[n * 33 + f]), 0, 0);
        __builtin_amdgcn_global_load_async_to_lds_b32(GASI(dsamp + d * 2048 + i),
                                                      LASI(&ds[n * 33 + f]), 0, 0);
    }
    __builtin_amdgcn_s_wait_asynccnt(0);
#else
    for (int i = tid; i < 2048; i += 256) {
        int n = i >> 5, f = i & 31;
        xe[n * 33 + f] = xenc[b * 2048 + i];
        ds[n * 33 + f] = dsamp[d * 2048 + i];
    }
#endif
    __builtin_prefetch(dist_para + tid * 512, 0, 1);  // dist_para is L2-resident
    __syncthreads();

    if (tid < 64) {
        float sx = 0.f, sy = 0.f;
        for (int f = 0; f < 32; ++f) {
            float a = xe[tid * 33 + f]; sx += a * a;
            float c = ds[tid * 33 + f]; sy += c * c;
        }
        x2[tid] = sx; y2[tid] = sy;
        int nn = num_node[b], nd = dict_nnode[d];
        float r = (tid < nn) ? 1.f : 0.f;
        float c = (tid < nd) ? 1.f : 0.f;
        rmk[tid] = r; wiL[tid] = r / (float)nn;
        cmk[tid] = c; wjL[tid] = c / (float)nd;
    }
    __syncthreads();

    // M[i,j] = x2[i] + y2[j] - 2 * <xe[i,:], ds[j,:]>   via WMMA (B transposed)
    for (int t = wave; t < 16; t += 8) {
        int m0 = (t >> 2) << 4, n0 = (t & 3) << 4;
        v8f acc = zero8();
        acc = wmma_tile_f32(xe, 33, ds, 33, m0, n0, 32, acc, true);
        const int lane = tid & 31, half = lane >> 4, col = lane & 15;
#pragma unroll
        for (int v = 0; v < 8; ++v) {
            int i = m0 + v + 8 * half, j = n0 + col;
            Mld[i * 65 + j] = x2[i] + y2[j] - 2.f * acc[v];
        }
    }
    __syncthreads();

    for (int l = 0; l < 8; ++l) {
        const float lam = LAM[l];
        for (int idx = tid; idx < 4096; idx += 256) {
            int i = idx >> 6, j = idx & 63;
            Kld[i * 65 + j] = expf(-lam * Mld[i * 65 + j]) * rmk[i] * cmk[j];
        }
        if (tid < 64) uiL[tid] = 1.f;
        __syncthreads();

        for (int it = 0; it < 10; ++it) {
            { // vi[j] = wj[j] / (K^T u)[j]
                int j = tid & 63, p = tid >> 6;
                float s = 0.f;
                for (int q = 0; q < 16; ++q) {
                    int i = p * 16 + q;
                    s += Kld[i * 65 + j] * uiL[i];
                }
                part[p * 64 + j] = s;
            }
            __syncthreads();
            if (tid < 64) {
                float s = part[tid] + part[64 + tid] + part[128 + tid] + part[192 + tid];
                viL[tid] = wjL[tid] / (s + EPSF);
            }
            __syncthreads();
            { // ui[i] = wi[i] / (K v)[i]
                int i = tid & 63, p = tid >> 6;
                float s = 0.f;
                for (int q = 0; q < 16; ++q) {
                    int j = p * 16 + q;
                    s += Kld[i * 65 + j] * viL[j];
                }
                part[p * 64 + i] = s;
            }
            __syncthreads();
            if (tid < 64) {
                float s = part[tid] + part[64 + tid] + part[128 + tid] + part[192 + tid];
                uiL[tid] = wiL[tid] / (s + EPSF);
            }
            __syncthreads();
        }

        { // row-wise min/max of G = ui*K*vi
            int i = tid & 63, p = tid >> 6;
            float mn = 3.4e38f, mx = -3.4e38f;
            for (int q = 0; q < 16; ++q) {
                int j = p * 16 + q;
                float g = uiL[i] * Kld[i * 65 + j] * viL[j];
                mn = fminf(mn, g);
                mx = fmaxf(mx, g);
            }
            pmin[p * 64 + i] = mn; pmax[p * 64 + i] = mx;
        }
        __syncthreads();
        if (tid < 64) {
            rmin[tid] = fminf(fminf(pmin[tid], pmin[64 + tid]),
                              fminf(pmin[128 + tid], pmin[192 + tid]));
            rmax[tid] = fmaxf(fmaxf(pmax[tid], pmax[64 + tid]),
                              fmaxf(pmax[128 + tid], pmax[192 + tid]));
        }
        __syncthreads();

        // GM = normalize(G) * M   (in place into Kld)
        for (int idx = tid; idx < 4096; idx += 256) {
            int i = idx >> 6, j = idx & 63;
            float g  = uiL[i] * Kld[i * 65 + j] * viL[j];
            float gn = (g - rmin[i]) / (rmax[i] - rmin[i] + EPSF);
            Kld[i * 65 + j] = gn * Mld[i * 65 + j];
        }
        __syncthreads();

        { // gd[.,.,c] = sum_{ij} GM[i,j] * dist_para[i*64+j, c]
            int c = tid & 31, g8 = tid >> 5;
            float s = 0.f;
            for (int q = 0; q < 8; ++q) {
                int i = g8 * 8 + q;
                for (int j = 0; j < 64; ++j)
                    s += Kld[i * 65 + j] * dist_para[(i * 64 + j) * 32 + c];
            }
            gpart[g8 * 32 + c] = s;
        }
        __syncthreads();
        if (tid < 32) {
            float s = 0.f;
            for (int g8 = 0; g8 < 8; ++g8) s += gpart[g8 * 32 + tid];
            gd[((l * 32 + b) * 30 + d) * 32 + tid] = s;
        }
        __syncthreads();
    }
}

// ---------------------------------------------------------------------------
// K5: softmax over lambda, weighted embed, fc1+fc2 head, output + kl
// ---------------------------------------------------------------------------
__global__ __launch_bounds__(256) void head_kernel(
    const float* __restrict__ gd, const float* __restrict__ wl,
    const float* __restrict__ fc1_w, const float* __restrict__ fc1_b,
    const float* __restrict__ fc2_w, const float* __restrict__ fc2_b,
    const float* __restrict__ klin, float* __restrict__ out)
{
    __shared__ float sc[256];    // [l][b]
    __shared__ float coef[256];  // [l][b]
    __shared__ float hL[2048];   // [b][o]
    const int tid = threadIdx.x;

    {
        int l = tid >> 5, b = tid & 31;
        const float* gp = gd + (l * 32 + b) * 960;
        float s = 0.f;
        for (int k = 0; k < 960; ++k) s += gp[k] * wl[k];
        sc[l * 32 + b] = s;
    }
    __syncthreads();
    if (tid < 32) {
        float mx = -3.4e38f;
        for (int l = 0; l < 8; ++l) mx = fmaxf(mx, sc[l * 32 + tid]);
        float e[8], ssum = 0.f;
        for (int l = 0; l < 8; ++l) { e[l] = expf(sc[l * 32 + tid] - mx); ssum += e[l]; }
        for (int l = 0; l < 8; ++l) coef[l * 32 + tid] = e[l] / ssum;
    }
    __syncthreads();

    // h[b,o] = b1[o] + sum_k (sum_l coef[l,b]*gd[l*32+b, k]) * fc1_w[o,k]
    for (int p = tid; p < 2048; p += 256) {
        int b = p >> 6, o = p & 63;
        float s = fc1_b[o];
        const float* w = fc1_w + o * 960;
        float cf[8];
#pragma unroll
        for (int l = 0; l < 8; ++l) cf[l] = coef[l * 32 + b];
        for (int k = 0; k < 960; ++k) {
            float e = 0.f;
#pragma unroll
            for (int l = 0; l < 8; ++l) e += cf[l] * gd[(l * 32 + b) * 960 + k];
            s += e * w[k];
        }
        hL[p] = s;
    }
    __syncthreads();
    if (tid < 96) {
        int b = tid / 3, cl = tid % 3;
        float s = fc2_b[cl];
        for (int o = 0; o < 64; ++o) s += hL[b * 64 + o] * fc2_w[cl * 64 + o];
        out[b * 3 + cl] = s;
    }
    if (tid == 0) out[96] = *klin;
}

// ---------------------------------------------------------------------------
extern "C" void kernel_launch(void* const* d_in, const int* in_sizes, int n_in,
                              void* d_out, int out_size, void* d_ws, size_t ws_size,
                              hipStream_t stream)
{
    const float* x          = (const float*)d_in[0];
    const float* adj        = (const float*)d_in[1];
    const int*   num_node   = (const int*)d_in[2];
    const float* dict_feat  = (const float*)d_in[3];
    const float* dict_adj   = (const float*)d_in[4];
    const int*   dict_nnode = (const int*)d_in[5];
    const float* eW1 = (const float*)d_in[6];   const float* eb1 = (const float*)d_in[7];
    const float* eW2 = (const float*)d_in[8];   const float* eb2 = (const float*)d_in[9];
    const float* eW3 = (const float*)d_in[10];  const float* eb3 = (const float*)d_in[11];
    const float* dW1 = (const float*)d_in[12];  const float* db1 = (const float*)d_in[13];
    const float* dW2 = (const float*)d_in[14];  const float* db2 = (const float*)d_in[15];
    const float* dW3 = (const float*)d_in[16];  const float* db3 = (const float*)d_in[17];
    const float* mlp_w  = (const float*)d_in[18]; const float* mlp_b  = (const float*)d_in[19];
    const float* mlp2_w = (const float*)d_in[20]; const float* mlp2_b = (const float*)d_in[21];
    const float* dist_para    = (const float*)d_in[22];
    const float* weight_lamda = (const float*)d_in[23];
    const float* fc1_w = (const float*)d_in[24]; const float* fc1_b = (const float*)d_in[25];
    const float* fc2_w = (const float*)d_in[26]; const float* fc2_b = (const float*)d_in[27];

    float* ws     = (float*)d_ws;
    float* x_enc  = ws;                 // 32*64*32 = 65536
    float* d_enc  = ws + 65536;         // 30*64*32 = 61440
    float* d_samp = ws + 126976;        // 61440
    float* gd     = ws + 188416;        // 256*30*32 = 245760
    float* kl     = ws + 434176;        // 1
    float* out    = (float*)d_out;      // 96 logits + kl

    gcn_kernel<<<32, 256, 0, stream>>>(x, adj, eW1, eb1, eW2, eb2, eW3, eb3, x_enc);
    gcn_kernel<<<30, 256, 0, stream>>>(dict_feat, dict_adj, dW1, db1, dW2, db2, dW3, db3, d_enc);
    attn_kernel<<<1, 256, 0, stream>>>(x_enc, d_enc, mlp_w, mlp_b, mlp2_w, mlp2_b, d_samp, kl);
    fuse_kernel<<<960, 256, 0, stream>>>(x_enc, d_samp, num_node, dict_nnode, dist_para, gd);
    head_kernel<<<1, 256, 0, stream>>>(gd, weight_lamda, fc1_w, fc1_b, fc2_w, fc2_b, kl, out);
}